// MessagePassing_15015205667393
// MI455X (gfx1250) — compile-verified
//
#include <hip/hip_runtime.h>
#include <math.h>

// Problem constants (fixed by setup_inputs): N=1, H=128, W=256, C=32.
#define Hh 128
#define Ww 256
#define Cc 32
#define HW (Hh * Ww)
#define HWC (Hh * Ww * Cc)

// LDS tile: 32 rows x 32 floats, row stride 33 dwords (TDM pad feature) ->
// conflict-free for BOTH row-wise and column-wise 32-lane reads.
#define ROWSTRIDE 33
#define SLOTF (32 * ROWSTRIDE)   // 1056 floats = 4224 B per tile
#define WAVES_PER_BLOCK 4

typedef unsigned int u32x4 __attribute__((ext_vector_type(4)));
typedef int          i32x4 __attribute__((ext_vector_type(4)));
typedef int          i32x8 __attribute__((ext_vector_type(8)));

__device__ __forceinline__ float readlane_f(float v, int l) {
  return __int_as_float(__builtin_amdgcn_readlane(__float_as_int(v), l));
}

__device__ __forceinline__ float wave_min32(float v) {
#pragma unroll
  for (int m = 16; m > 0; m >>= 1) v = fminf(v, __shfl_xor(v, m, 32));
  return v;
}
__device__ __forceinline__ float wave_max32(float v) {
#pragma unroll
  for (int m = 16; m > 0; m >>= 1) v = fmaxf(v, __shfl_xor(v, m, 32));
  return v;
}
__device__ __forceinline__ float wave_sum32(float v) {
#pragma unroll
  for (int m = 16; m > 0; m >>= 1) v += __shfl_xor(v, m, 32);
  return v;
}

// Issue one TDM DMA: 32x32 f32 tile from global -> LDS, with 1-dword pad
// after every 32 dwords (row stride becomes 33 dwords in LDS).
// D# layout per CDNA5 ISA ch.8 (group0: count/lds/global/type; group1:
// data_size=4B, pad_enable, pad_interval=32dw, pad_amount=1dw,
// tensor_dim0=tensor_dim1=tile_dim0=tile_dim1=32, dim0_stride=32).
__device__ __forceinline__ void tdm_load_tile32x32(unsigned lds_byte_off,
                                                   const float* gptr) {
  unsigned long long ga = (unsigned long long)(uintptr_t)gptr;
  u32x4 g0;
  g0[0] = 1u;                                        // count=1, user mode
  g0[1] = lds_byte_off;                              // lds_addr
  g0[2] = (unsigned)(ga & 0xFFFFFFFFu);              // global_addr[31:0]
  g0[3] = (unsigned)((ga >> 32) & 0x01FFFFFFu)       // global_addr[56:32]
          | (2u << 30);                              // type=2 (image)
  i32x8 g1;
  g1[0] = (int)((2u << 16)      // data_size = 4 bytes
              | (1u << 20)      // pad_enable
              | (4u << 22)      // pad_interval: 32 dwords
              | (0u << 25));    // pad_amount: 1 dword
  g1[1] = (int)(32u << 16);     // tensor_dim0[15:0]=32 (in [31:16])
  g1[2] = (int)(32u << 16);     // tensor_dim0 hi=0 | tensor_dim1 lo=32
  g1[3] = (int)(32u << 16);     // tensor_dim1 hi=0 | tile_dim0=32
  g1[4] = (int)(32u);           // tile_dim1=32 | tile_dim2=0
  g1[5] = 32;                   // tensor_dim0_stride lo32 = 32 elements
  g1[6] = 0;                    // stride hi | tensor_dim1_stride lo
  g1[7] = 0;                    // tensor_dim1_stride hi
  i32x4 z4 = {0, 0, 0, 0};      // groups 2/3 unused (<=2D tensor)
  i32x8 z8 = {0, 0, 0, 0, 0, 0, 0, 0};
  // clang-23 / therock-10.0 lane: 6-arg builtin form (probe table):
  // (u32x4 g0, i32x8 g1, i32x4, i32x4, i32x8, i32 cpol)
  __builtin_amdgcn_tensor_load_to_lds(g0, g1, z4, z4, z8, 0);
}

// One BP iteration. One wave owns pixel (h,w) and its two jump matrices:
//   J0 = jump[0][h][w] -> msg_out[0] at (h,w+1)  (plain)   [if w<W-1]
//                         msg_out[1] at (h,w)    (transposed)
//   J1 = jump[1][h][w] -> msg_out[2] at (h+1,w)  (plain)   [if h<H-1]
//                         msg_out[3] at (h,w)    (transposed)
// Each jump matrix is streamed from HBM exactly once per iteration.
__global__ __launch_bounds__(WAVES_PER_BLOCK * 32) void bp_iter_kernel(
    const float* __restrict__ prob, const float* __restrict__ edge,
    const float* __restrict__ msg_in, const float* __restrict__ jump,
    float* __restrict__ msg_out) {
  __shared__ float smem[WAVES_PER_BLOCK * 2 * SLOTF];

  const int lane = threadIdx.x & 31;
  const int wid  = threadIdx.x >> 5;
  const int p    = blockIdx.x * WAVES_PER_BLOCK + wid;
  const int h    = p / Ww;
  const int w    = p % Ww;
  const bool hvalid = (w < Ww - 1);
  const bool vvalid = (h < Hh - 1);

  float* slot0 = &smem[(wid * 2 + 0) * SLOTF];
  float* slot1 = &smem[(wid * 2 + 1) * SLOTF];
  // Low 32 bits of a generic LDS pointer are the LDS byte offset (aperture
  // mapping, CDNA5 ISA 10.2).
  const unsigned lds0 = (unsigned)(uintptr_t)slot0;
  const unsigned lds1 = (unsigned)(uintptr_t)slot1;

  // Kick off both tile DMAs (in-bounds even at boundaries; results unused
  // there). TDM ignores EXEC; descriptors are wave-uniform.
  const float* j0 = jump + (size_t)(h * Ww + w) * (Cc * Cc);
  const float* j1 = jump + (size_t)(HW + h * Ww + w) * (Cc * Cc);
  tdm_load_tile32x32(lds0, j0);
  tdm_load_tile32x32(lds1, j1);
  asm volatile("" ::: "memory");

  // Overlap DMA with the (coalesced) gathers for the tmp vectors.
  const int bh = (h * Ww + w) * Cc + lane;
  const float m0 = msg_in[0 * HWC + bh];
  const float m1 = msg_in[1 * HWC + bh];
  const float m2 = msg_in[2 * HWC + bh];
  const float m3 = msg_in[3 * HWC + bh];
  const float tot  = -prob[bh] + (m0 + m1 + m2 + m3);
  const float tmpA = tot - m1;                 // sender (h,w), dir 0
  const float tmpC = tot - m3;                 // sender (h,w), dir 2
  const float eA = edge[0 * HW + h * Ww + w];
  const float eC = edge[2 * HW + h * Ww + w];

  float tmpB = 0.f, eB = 0.f, tmpD = 0.f, eD = 0.f;
  if (hvalid) {  // sender (h,w+1), dir 1 (uses J0^T)
    const int br = (h * Ww + (w + 1)) * Cc + lane;
    const float r1 = msg_in[1 * HWC + br];
    const float r2 = msg_in[2 * HWC + br];
    const float r3 = msg_in[3 * HWC + br];
    tmpB = -prob[br] + (r1 + r2 + r3);         // total - msg[opp(1)=0]
    eB   = edge[1 * HW + h * Ww + (w + 1)];
  }
  if (vvalid) {  // sender (h+1,w), dir 3 (uses J1^T)
    const int bd = ((h + 1) * Ww + w) * Cc + lane;
    const float d0 = msg_in[0 * HWC + bd];
    const float d1 = msg_in[1 * HWC + bd];
    const float d3 = msg_in[3 * HWC + bd];
    tmpD = -prob[bd] + (d0 + d1 + d3);         // total - msg[opp(3)=2]
    eD   = edge[3 * HW + (h + 1) * Ww + w];
  }

  // ---- horizontal pair: needs only the first TDM op (in-order per wave)
  asm volatile("s_wait_tensorcnt 0x1" ::: "memory");
  if (hvalid) {
    float mr = 1e30f, ml = 1e30f;
#pragma unroll
    for (int k = 0; k < 32; ++k) {
      const float a  = readlane_f(tmpA, k);          // SGPR broadcast
      const float b  = readlane_f(tmpB, k);
      const float jc = slot0[k * ROWSTRIDE + lane];  // J0[k][lane] (column)
      const float jr = slot0[lane * ROWSTRIDE + k];  // J0[lane][k] (row)
      mr = fminf(mr, fmaf(eA, jc, a));               // dir0 -> (h,w+1)
      ml = fminf(ml, fmaf(eB, jr, b));               // dir1 -> (h,w)
    }
    mr -= wave_min32(mr);
    ml -= wave_min32(ml);
    msg_out[0 * HWC + (h * Ww + (w + 1)) * Cc + lane] = mr;
    msg_out[1 * HWC + (h * Ww + w) * Cc + lane]       = ml;
  } else {  // w == W-1: write this row's boundary-zero messages
    msg_out[0 * HWC + (h * Ww + 0) * Cc + lane]        = 0.f;
    msg_out[1 * HWC + (h * Ww + (Ww - 1)) * Cc + lane] = 0.f;
  }

  // ---- vertical pair
  asm volatile("s_wait_tensorcnt 0x0" ::: "memory");
  if (vvalid) {
    float md = 1e30f, mu = 1e30f;
#pragma unroll
    for (int k = 0; k < 32; ++k) {
      const float c  = readlane_f(tmpC, k);
      const float d  = readlane_f(tmpD, k);
      const float jc = slot1[k * ROWSTRIDE + lane];  // J1[k][lane]
      const float jr = slot1[lane * ROWSTRIDE + k];  // J1[lane][k]
      md = fminf(md, fmaf(eC, jc, c));               // dir2 -> (h+1,w)
      mu = fminf(mu, fmaf(eD, jr, d));               // dir3 -> (h,w)
    }
    md -= wave_min32(md);
    mu -= wave_min32(mu);
    msg_out[2 * HWC + ((h + 1) * Ww + w) * Cc + lane] = md;
    msg_out[3 * HWC + (h * Ww + w) * Cc + lane]       = mu;
  } else {  // h == H-1: write this column's boundary-zero messages
    msg_out[2 * HWC + (0 * Ww + w) * Cc + lane]        = 0.f;
    msg_out[3 * HWC + ((Hh - 1) * Ww + w) * Cc + lane] = 0.f;
  }
}

// beliefs = cost + sum(msg); output = softmax(-beliefs) = softmax(prob - summsg)
__global__ __launch_bounds__(256) void belief_softmax_kernel(
    const float* __restrict__ prob, const float* __restrict__ msg,
    float* __restrict__ out) {
  const int lane = threadIdx.x & 31;
  const int wid  = threadIdx.x >> 5;
  const int p    = blockIdx.x * 8 + wid;
  const int idx  = p * Cc + lane;
  const float v = prob[idx] - (msg[idx] + msg[HWC + idx] +
                               msg[2 * HWC + idx] + msg[3 * HWC + idx]);
  const float mx = wave_max32(v);
  const float e  = __expf(v - mx);
  const float s  = wave_sum32(e);
  out[idx] = e / s;
}

extern "C" void kernel_launch(void* const* d_in, const int* in_sizes, int n_in,
                              void* d_out, int out_size, void* d_ws,
                              size_t ws_size, hipStream_t stream) {
  const float* prob = (const float*)d_in[0];  // (1,H,W,C)
  const float* edge = (const float*)d_in[1];  // (1,4,H,W)
  const float* msg0 = (const float*)d_in[2];  // (1,4,H,W,C), zeros
  const float* jump = (const float*)d_in[3];  // (2,H,W,C,C)

  float* A = (float*)d_ws;            // 4*H*W*C floats = 16 MB
  float* B = A + (size_t)4 * HWC;     // another 16 MB

  const dim3 blk(WAVES_PER_BLOCK * 32);
  const dim3 grd(HW / WAVES_PER_BLOCK);
  bp_iter_kernel<<<grd, blk, 0, stream>>>(prob, edge, msg0, jump, A);
  bp_iter_kernel<<<grd, blk, 0, stream>>>(prob, edge, A, jump, B);
  bp_iter_kernel<<<grd, blk, 0, stream>>>(prob, edge, B, jump, A);
  belief_softmax_kernel<<<HW / 8, 256, 0, stream>>>(prob, A, (float*)d_out);
}